// QuantumFluxAttention_14963666059655
// MI455X (gfx1250) — compile-verified
//
#include <hip/hip_runtime.h>
#include <hip/hip_bf16.h>

// QuantumFluxAttention for MI455X (gfx1250, wave32).
//
// sim[b,i,j] = x_i*u_j + y_i*v_j  with  u_j = x_j + a*y_j,  v_j = y_j - a*x_j
// gate = 1/(eps+|sim|); att = (gate>0.5) ? sim*gate : 0
// ctx[b,i] = sum_j att[i,j] * s_j
//
// Store-bound: 1 GiB of attention -> ~46us floor @ 23.3 TB/s.
// V_WMMA_F32_16X16X4_F32 computes each 16x16 sim tile in one instruction.
// Mapping M=j, N=i makes each lane own 8 *consecutive* j's of one attention
// row -> two coalesced nontemporal global_store_b128 per tile per lane.
// Context operands come from 4x global_load_b128 of cache-resident states
// (replaces 16 ds_bpermute/iter from the previous revision).

typedef __attribute__((ext_vector_type(2))) float v2f;
typedef __attribute__((ext_vector_type(4))) float v4f;
typedef __attribute__((ext_vector_type(8))) float v8f;

#define BATCH 4
#define NSEQ  8192
#define NT    (NSEQ / 16)     // 512 j-tiles per row-stripe
#define WAVES 8               // waves per block (256 threads)

#define EPS   1e-5f
#define ALPHA 0.1f

__global__ __launch_bounds__(256) void
qfa_kernel(const float* __restrict__ states, float* __restrict__ out)
{
    const int tid   = threadIdx.x;
    const int wave  = tid >> 5;
    const int lane  = tid & 31;
    const int lhalf = lane >> 4;   // 0: lanes 0-15, 1: lanes 16-31
    const int lidx  = lane & 15;

    // one block per (batch, 16-row i-stripe)
    const int stripe = blockIdx.x;          // 0 .. BATCH*NT-1
    const int b      = stripe >> 9;         // / NT
    const int it     = stripe & (NT - 1);
    const int i0     = it << 4;

    const float* __restrict__ S = states + (size_t)b * NSEQ * 2;
    float* __restrict__ att_out = out + (size_t)b * NSEQ * NSEQ;
    float* __restrict__ ctx_out = out + (size_t)BATCH * NSEQ * NSEQ;

    // B operand (loop-invariant): B[K,N=i]. Lanes 0-15 hold K=0,1 rows
    // ({x_i, y_i} for column i = i0+lidx); lanes 16-31 hold K=2,3 (zero).
    v2f bop;
    {
        float xi = S[(i0 + lidx) * 2 + 0];
        float yi = S[(i0 + lidx) * 2 + 1];
        bop.x = lhalf ? 0.0f : xi;
        bop.y = lhalf ? 0.0f : yi;
    }

    float cx = 0.0f, cy = 0.0f;   // per-lane context partial for column i0+lidx

    for (int jt = wave; jt < NT; jt += WAVES) {
        const int j0 = jt << 4;

        // A operand source: s_{j0+lidx} per lane (lanes 16-31 zeroed below).
        float xj = S[(j0 + lidx) * 2 + 0];
        float yj = S[(j0 + lidx) * 2 + 1];
        float u  = fmaf( ALPHA, yj, xj);   // x_j + a*y_j
        float v  = fmaf(-ALPHA, xj, yj);   // y_j - a*x_j

        // A operand: A[M=j,K]. Lanes 0-15: VGPR0=K0(u), VGPR1=K1(v) for row
        // j = j0+lidx; lanes 16-31 supply K=2,3 (zero).
        v2f a;
        a.x = lhalf ? 0.0f : u;
        a.y = lhalf ? 0.0f : v;

        // Context source data: the 8 (x,y) pairs this lane reduces against,
        // s[j0+8*lhalf .. +7] = 64 contiguous bytes (64B aligned), identical
        // across each half-wave -> 4 cache-hit b128 loads, no lane shuffles.
        const v4f* __restrict__ cs =
            (const v4f*)(S + (size_t)(j0 + (lhalf << 3)) * 2);
        v4f s01 = cs[0];   // x[j+0] y[j+0] x[j+1] y[j+1]
        v4f s23 = cs[1];
        v4f s45 = cs[2];
        v4f s67 = cs[3];

        v8f c = {};
        // 8 args: (neg_a, A, neg_b, B, c_mod, C, reuse_a, reuse_b)
        c = __builtin_amdgcn_wmma_f32_16x16x4_f32(
                false, a, false, bop, (short)0, c, false, false);

        // Lane holds D[M = r + 8*lhalf, N = lidx] = sim(i = i0+lidx,
        // j = j0 + 8*lhalf + r): 8 consecutive j per lane.
        float att[8];
#pragma unroll
        for (int r = 0; r < 8; ++r) {
            float sim = c[r];
            float g   = __builtin_amdgcn_rcpf(EPS + __builtin_fabsf(sim));
            att[r]    = (g > 0.5f) ? sim * g : 0.0f;
        }

        // Two coalesced 16B nontemporal stores of one attention row segment.
        float* p = att_out + (size_t)(i0 + lidx) * NSEQ + j0 + (lhalf << 3);
        v4f o0 = { att[0], att[1], att[2], att[3] };
        v4f o1 = { att[4], att[5], att[6], att[7] };
        __builtin_nontemporal_store(o0, (v4f*)p);
        __builtin_nontemporal_store(o1, (v4f*)(p + 4));

        // Context: ctx_i += sum_r att[r] * s_{j0 + 8*lhalf + r}.
        cx = fmaf(att[0], s01.x, cx);  cy = fmaf(att[0], s01.y, cy);
        cx = fmaf(att[1], s01.z, cx);  cy = fmaf(att[1], s01.w, cy);
        cx = fmaf(att[2], s23.x, cx);  cy = fmaf(att[2], s23.y, cy);
        cx = fmaf(att[3], s23.z, cx);  cy = fmaf(att[3], s23.w, cy);
        cx = fmaf(att[4], s45.x, cx);  cy = fmaf(att[4], s45.y, cy);
        cx = fmaf(att[5], s45.z, cx);  cy = fmaf(att[5], s45.w, cy);
        cx = fmaf(att[6], s67.x, cx);  cy = fmaf(att[6], s67.y, cy);
        cx = fmaf(att[7], s67.z, cx);  cy = fmaf(att[7], s67.w, cy);
    }

    // Combine the two wave halves (same i, disjoint j sets).
    cx += __shfl_xor(cx, 16, 32);
    cy += __shfl_xor(cy, 16, 32);

    // Deterministic cross-wave reduction of context partials via LDS.
    __shared__ float red[WAVES][16][2];
    if (lhalf == 0) {
        red[wave][lidx][0] = cx;
        red[wave][lidx][1] = cy;
    }
    __syncthreads();
    if (tid < 16) {
        float sx = 0.0f, sy = 0.0f;
#pragma unroll
        for (int w = 0; w < WAVES; ++w) {
            sx += red[w][tid][0];
            sy += red[w][tid][1];
        }
        ctx_out[(size_t)(b * NSEQ + i0 + tid) * 2 + 0] = sx;
        ctx_out[(size_t)(b * NSEQ + i0 + tid) * 2 + 1] = sy;
    }
}

extern "C" void kernel_launch(void* const* d_in, const int* in_sizes, int n_in,
                              void* d_out, int out_size, void* d_ws, size_t ws_size,
                              hipStream_t stream)
{
    (void)in_sizes; (void)n_in; (void)out_size; (void)d_ws; (void)ws_size;
    const float* states = (const float*)d_in[0];
    float* out = (float*)d_out;

    dim3 grid(BATCH * NT);   // 2048 blocks, 8 waves each
    dim3 block(256);
    qfa_kernel<<<grid, block, 0, stream>>>(states, out);
}